// Net1_44598940402302
// MI455X (gfx1250) — compile-verified
//
#include <hip/hip_runtime.h>
#include <hip/hip_bf16.h>
#include <math.h>

// ---------------------------------------------------------------------------
// 2-layer GCN for MI455X (gfx1250, wave32).
//   h1   = x @ W1                       (WMMA f32 16x16x4, K=512)
//   a1   = D^-1/2 A D^-1/2 h1 + b1 ; relu
//   lg   = a1 @ W2                      (WMMA f32 16x16x4, K=16)
//   out  = log_softmax(D^-1/2 A D^-1/2 lg + b2)
//
// Cache policy: x (205 MB) is a zero-reuse stream -> non-temporal loads so it
// does not evict the reused graph tables (h/agg: 6.4 MB, edge list: 102 MB)
// from the 192 MB L2, which the atomic edge-aggregation phase depends on.
// ---------------------------------------------------------------------------

typedef __attribute__((ext_vector_type(2))) float v2f;
typedef __attribute__((ext_vector_type(8))) float v8f;

#define D_IN   512
#define D_HID  16
#define LDS_STRIDE1 (D_IN + 4)   // padded transpose of W1: conflict-free ds_load_b64
#define LDS_STRIDE2 (D_HID + 4)  // padded transpose of W2

// ---------------- degree / normalization ----------------------------------

__global__ void k_init_deg(float* deg, int nNodes) {
    int i = blockIdx.x * blockDim.x + threadIdx.x;
    if (i < nNodes) deg[i] = 1.0f;               // self-loop contribution
}

__global__ void k_count_deg(const long long* __restrict__ ei, float* deg,
                            long long nEdges) {
    long long e = (long long)blockIdx.x * blockDim.x + threadIdx.x;
    if (e < nEdges) {
        int d = (int)ei[nEdges + e];             // dst row
        atomicAdd(&deg[d], 1.0f);
    }
}

__global__ void k_dinv(float* deg, int nNodes) {  // in-place deg -> deg^-1/2
    int i = blockIdx.x * blockDim.x + threadIdx.x;
    if (i < nNodes) deg[i] = rsqrtf(deg[i]);      // deg >= 1 always
}

// ---------------- GEMM1: h1 = x @ W1 (N x 512 x 16), WMMA f32 16x16x4 ------

__global__ __launch_bounds__(256) void k_gemm1_wmma(
        const float* __restrict__ x, const float* __restrict__ W1,
        float* __restrict__ h1, int nNodes) {
    __shared__ float w1t[D_HID * LDS_STRIDE1];    // W1 transposed: w1t[col][k]
    const int t = threadIdx.x;

    // cooperative transpose of W1 (512x16, row-major) into LDS
    #pragma unroll
    for (int i = 0; i < (D_IN * D_HID) / 256; ++i) {
        int idx = i * 256 + t;                    // coalesced global read
        int k = idx >> 4, c = idx & 15;
        w1t[c * LDS_STRIDE1 + k] = W1[idx];
    }
    __syncthreads();

    const int wave = t >> 5;
    const int lane = t & 31;
    const int rowBase = (blockIdx.x * 8 + wave) * 16;   // one 16-row tile/wave
    if (rowBase >= nNodes) return;                // N % 16 == 0: whole tile

    const int m  = lane & 15;                     // A row / B col / D col
    const int kh = (lane >> 4) * 2;               // K sub-offset (0 or 2)
    const float* aRow = x + (size_t)(rowBase + m) * D_IN + kh;
    const float* bRow = &w1t[m * LDS_STRIDE1 + kh];

    v8f acc = {};
    for (int k0 = 0; k0 < D_IN; k0 += 32) {       // 32 floats = one 128B line
        __builtin_prefetch(aRow + k0 + 256, 0, 0);        // 1KB ahead -> GL2
        #pragma unroll
        for (int kk = 0; kk < 32; kk += 4) {
            v2f a = __builtin_nontemporal_load((const v2f*)(aRow + k0 + kk));
            v2f b = *(const v2f*)(bRow + k0 + kk);         // ds_load_b64
            acc = __builtin_amdgcn_wmma_f32_16x16x4_f32(
                      false, a, false, b, (short)0, acc, false, false);
        }
    }

    // D: VGPR r -> row (r + (lane>=16 ? 8 : 0)), col = lane&15
    const int rOff = (lane >> 4) * 8;
    float* outBase = h1 + (size_t)rowBase * D_HID + m;
    #pragma unroll
    for (int r = 0; r < 8; ++r)
        outBase[(size_t)(r + rOff) * D_HID] = acc[r];
}

// ---------------- GEMM2: logits = h2 @ W2 (N x 16 x 16) --------------------

__global__ __launch_bounds__(256) void k_gemm2_wmma(
        const float* __restrict__ h2, const float* __restrict__ W2,
        float* __restrict__ lg, int nNodes) {
    __shared__ float w2t[D_HID * LDS_STRIDE2];
    const int t = threadIdx.x;
    {   // 16x16 transpose, one element per thread
        int k = t >> 4, c = t & 15;
        w2t[c * LDS_STRIDE2 + k] = W2[t];
    }
    __syncthreads();

    const int wave = t >> 5;
    const int lane = t & 31;
    const int rowBase = (blockIdx.x * 8 + wave) * 16;
    if (rowBase >= nNodes) return;

    const int m  = lane & 15;
    const int kh = (lane >> 4) * 2;
    const float* aRow = h2 + (size_t)(rowBase + m) * D_HID + kh;
    const float* bRow = &w2t[m * LDS_STRIDE2 + kh];

    v8f acc = {};
    #pragma unroll
    for (int k0 = 0; k0 < D_HID; k0 += 4) {
        v2f a = *(const v2f*)(aRow + k0);         // h2 is hot in L2: RT loads
        v2f b = *(const v2f*)(bRow + k0);
        acc = __builtin_amdgcn_wmma_f32_16x16x4_f32(
                  false, a, false, b, (short)0, acc, false, false);
    }

    const int rOff = (lane >> 4) * 8;
    float* outBase = lg + (size_t)rowBase * D_HID + m;
    #pragma unroll
    for (int r = 0; r < 8; ++r)
        outBase[(size_t)(r + rOff) * D_HID] = acc[r];
}

// ---------------- aggregation --------------------------------------------

// agg[i][f] = h[i][f] * dinv[i]^2   (self-loop term; also fully initializes agg)
__global__ void k_self_init(const float* __restrict__ h,
                            const float* __restrict__ dinv,
                            float* __restrict__ agg, int nNodes) {
    int tid = blockIdx.x * blockDim.x + threadIdx.x;
    int i = tid >> 4;
    if (i >= nNodes) return;
    float di = dinv[i];
    agg[tid] = h[tid] * di * di;
}

// agg[dst][f] += h[src][f] * dinv[src]*dinv[dst]  over all edges
// 16 lanes per edge: 64B-contiguous gather of h[src], L2 atomics into agg[dst]
__global__ void k_edge_agg(const long long* __restrict__ ei,
                           const float* __restrict__ dinv,
                           const float* __restrict__ h,
                           float* __restrict__ agg, long long nEdges) {
    long long tid = (long long)blockIdx.x * blockDim.x + threadIdx.x;
    long long e = tid >> 4;
    int f = (int)(tid & 15);
    if (e >= nEdges) return;
    int s = (int)ei[e];
    int d = (int)ei[nEdges + e];
    float nrm = dinv[s] * dinv[d];
    atomicAdd(&agg[(size_t)d * D_HID + f], h[(size_t)s * D_HID + f] * nrm);
}

__global__ void k_relu_bias(float* __restrict__ a, const float* __restrict__ b,
                            long long n) {
    long long tid = (long long)blockIdx.x * blockDim.x + threadIdx.x;
    if (tid < n) a[tid] = fmaxf(a[tid] + b[tid & 15], 0.0f);
}

// ---------------- log_softmax (in place on d_out) --------------------------

__global__ void k_logsoftmax(float* __restrict__ out,
                             const float* __restrict__ b2, int nNodes) {
    int i = blockIdx.x * blockDim.x + threadIdx.x;
    if (i >= nNodes) return;
    float v[D_HID];
    float mx = -INFINITY;
    #pragma unroll
    for (int f = 0; f < D_HID; ++f) {
        v[f] = out[(size_t)i * D_HID + f] + b2[f];
        mx = fmaxf(mx, v[f]);
    }
    float s = 0.0f;
    #pragma unroll
    for (int f = 0; f < D_HID; ++f) s += expf(v[f] - mx);
    float ls = logf(s);
    #pragma unroll
    for (int f = 0; f < D_HID; ++f)
        out[(size_t)i * D_HID + f] = v[f] - mx - ls;
}

// ---------------- launcher -------------------------------------------------

extern "C" void kernel_launch(void* const* d_in, const int* in_sizes, int n_in,
                              void* d_out, int out_size, void* d_ws, size_t ws_size,
                              hipStream_t stream) {
    const float*     x   = (const float*)d_in[0];
    const float*     W1  = (const float*)d_in[1];
    const float*     b1  = (const float*)d_in[2];
    const float*     W2  = (const float*)d_in[3];
    const float*     b2  = (const float*)d_in[4];
    const long long* ei  = (const long long*)d_in[5];   // int64 [2, E]

    const int       N = in_sizes[0] / D_IN;             // 100000
    const long long E = (long long)in_sizes[5] / 2;     // 6400000
    float* out = (float*)d_out;

    // workspace partition: dinv | bufA (h1/logits) | bufB (agg1/h2)
    float* dinv = (float*)d_ws;
    size_t npad = ((size_t)N + 255) & ~(size_t)255;
    float* bufA = dinv + npad;
    float* bufB = bufA + (size_t)N * D_HID;

    const int B = 256;
    int gN    = (N + B - 1) / B;
    int gE    = (int)((E + B - 1) / B);
    int gEf   = (int)((E * D_HID + B - 1) / B);
    int gNf   = (int)(((long long)N * D_HID + B - 1) / B);
    int tiles = (N + 15) / 16;
    int gGemm = (tiles + 7) / 8;                        // 8 waves/block

    // normalization: deg -> dinv (in place)
    k_init_deg <<<gN,  B, 0, stream>>>(dinv, N);
    k_count_deg<<<gE,  B, 0, stream>>>(ei, dinv, E);
    k_dinv     <<<gN,  B, 0, stream>>>(dinv, N);

    // layer 1
    k_gemm1_wmma<<<gGemm, B, 0, stream>>>(x, W1, bufA, N);
    k_self_init <<<gNf,  B, 0, stream>>>(bufA, dinv, bufB, N);
    k_edge_agg  <<<gEf,  B, 0, stream>>>(ei, dinv, bufA, bufB, E);
    k_relu_bias <<<gNf,  B, 0, stream>>>(bufB, b1, (long long)N * D_HID);

    // layer 2 (agg accumulates directly into d_out)
    k_gemm2_wmma<<<gGemm, B, 0, stream>>>(bufB, W2, bufA, N);
    k_self_init <<<gNf,  B, 0, stream>>>(bufA, dinv, out, N);
    k_edge_agg  <<<gEf,  B, 0, stream>>>(ei, dinv, bufA, out, E);
    k_logsoftmax<<<gN,   B, 0, stream>>>(out, b2, N);
}